// Feature_extraction_layer_86526411145632
// MI455X (gfx1250) — compile-verified
//
#include <hip/hip_runtime.h>
#include <hip/hip_bf16.h>

// ---------------------------------------------------------------------------
// Problem constants (fixed by setup_inputs)
// ---------------------------------------------------------------------------
#define BATCH 4
#define CCH   4
#define NP    16384   // P
#define NS    8192    // S
#define KNN   16

typedef float v2f __attribute__((ext_vector_type(2)));
typedef float v8f __attribute__((ext_vector_type(8)));

// workspace layout (bytes)
#define OFF_IDX    ((size_t)0)                         // BATCH*NS*KNN int  = 2 MB
#define OFF_STATS  ((size_t)(BATCH*NS*KNN*4))          // BATCH*64*2 float  = 2 KB
#define OFF_POOLED (OFF_STATS + (size_t)4096)          // BATCH*64*NS float = 8 MB

__device__ __forceinline__ float leakyf(float v, float s) {
    return v >= 0.0f ? v : s * v;
}

// ---------------------------------------------------------------------------
// K1: brute-force KNN.  One thread per query s; candidates staged in LDS.
//     pd = 2*q.p - |q|^2 - |p|^2  (matches reference formula; self -> 0 max).
//     Sorted descending top-16 maintained in registers via rare-branch insert.
// ---------------------------------------------------------------------------
#define KNN_CHUNK 2048
__global__ __launch_bounds__(256)
void knn_kernel(const float* __restrict__ x, const int* __restrict__ perm,
                int* __restrict__ idx_out) {
    __shared__ float lx[KNN_CHUNK], ly[KNN_CHUNK], lz[KNN_CHUNK], ln[KNN_CHUNK];
    const int b = blockIdx.y;
    const int s = blockIdx.x * 256 + threadIdx.x;
    const float* xb = x + (size_t)b * CCH * NP;

    const int qp = perm[b * NS + s];
    const float qx = xb[qp], qy = xb[NP + qp], qz = xb[2 * NP + qp];
    const float qn = qx * qx + qy * qy + qz * qz;

    float best[KNN];
    int   bidx[KNN];
#pragma unroll
    for (int j = 0; j < KNN; ++j) { best[j] = -3.4e38f; bidx[j] = 0; }

    for (int p0 = 0; p0 < NP; p0 += KNN_CHUNK) {
        __syncthreads();
        for (int j = threadIdx.x; j < KNN_CHUNK; j += 256) {
            float px = xb[p0 + j];
            float py = xb[NP + p0 + j];
            float pz = xb[2 * NP + p0 + j];
            lx[j] = px; ly[j] = py; lz[j] = pz;
            ln[j] = px * px + py * py + pz * pz;
        }
        __syncthreads();
        for (int j = 0; j < KNN_CHUNK; ++j) {
            float dot = qx * lx[j] + qy * ly[j] + qz * lz[j];
            float pd  = 2.0f * dot - qn - ln[j];
            if (pd > best[KNN - 1]) {           // rare after warm-up
                int p = p0 + j;
#pragma unroll
                for (int t = KNN - 1; t >= 1; --t) {
                    bool gt   = pd > best[t];
                    bool here = pd <= best[t - 1];
                    float nb = gt ? (here ? pd : best[t - 1]) : best[t];
                    int   ni = gt ? (here ? p  : bidx[t - 1]) : bidx[t];
                    best[t] = nb; bidx[t] = ni;
                }
                if (pd > best[0]) { best[0] = pd; bidx[0] = p; }
            }
        }
    }
    int* op = idx_out + ((size_t)(b * NS + s)) * KNN;
#pragma unroll
    for (int j = 0; j < KNN; ++j) op[j] = bidx[j];
}

// ---------------------------------------------------------------------------
// K2: GroupNorm statistics. Each 128-thread block owns 128 (s,k) slots,
//     computes y1 (conv path, ch 0..31) and y2 (pos path, ch 32..63) per slot,
//     then 64 threads reduce rows and atomically add (sum, sumsq) per channel.
// ---------------------------------------------------------------------------
#define ST_SLOTS 128
__global__ __launch_bounds__(ST_SLOTS)
void stats_kernel(const float* __restrict__ x, const int* __restrict__ idx,
                  const float* __restrict__ conv_w, const float* __restrict__ pos_w,
                  float* __restrict__ stats) {
    __shared__ float ytile[64][ST_SLOTS + 1];
    __shared__ float wconv[32 * 4];
    __shared__ float wpos[32 * 10];
    const int b   = blockIdx.y;
    const int g   = blockIdx.x * ST_SLOTS + threadIdx.x;   // slot in [0, NS*KNN)
    const int s   = g >> 4;
    const int k   = g & 15;

    for (int j = threadIdx.x; j < 128; j += ST_SLOTS) wconv[j] = conv_w[j];
    for (int j = threadIdx.x; j < 320; j += ST_SLOTS) wpos[j]  = pos_w[j];
    __syncthreads();

    const float* xb = x + (size_t)b * CCH * NP;
    const int*   ib = idx + ((size_t)(b * NS + s)) * KNN;
    const int pk = ib[k];
    const int p0 = ib[0];

    float x1c[4];
#pragma unroll
    for (int c = 0; c < 4; ++c) x1c[c] = xb[c * NP + pk];
    float pc0 = xb[p0], pc1 = xb[NP + p0], pc2 = xb[2 * NP + p0];
    float d0 = x1c[0] - pc0, d1 = x1c[1] - pc1, d2 = x1c[2] - pc2;
    float pemb[10] = { x1c[0], x1c[1], x1c[2], pc0, pc1, pc2,
                       d0, d1, d2, sqrtf(d0 * d0 + d1 * d1 + d2 * d2) };

#pragma unroll 4
    for (int o = 0; o < 32; ++o) {
        float y = 0.0f;
#pragma unroll
        for (int c = 0; c < 4; ++c) y += wconv[o * 4 + c] * x1c[c];
        ytile[o][threadIdx.x] = y;
    }
#pragma unroll 2
    for (int o = 0; o < 32; ++o) {
        float y = 0.0f;
#pragma unroll
        for (int c = 0; c < 10; ++c) y += wpos[o * 10 + c] * pemb[c];
        ytile[32 + o][threadIdx.x] = y;
    }
    __syncthreads();

    if (threadIdx.x < 64) {
        int ch = threadIdx.x;
        float sum = 0.0f, sq = 0.0f;
        for (int j = 0; j < ST_SLOTS; ++j) {
            float v = ytile[ch][j];
            sum += v; sq += v * v;
        }
        atomicAdd(&stats[(b * 64 + ch) * 2 + 0], sum);
        atomicAdd(&stats[(b * 64 + ch) * 2 + 1], sq);
    }
}

// ---------------------------------------------------------------------------
// K3: fused finalize with WMMA attention MLP.
//     Block = 128 threads = 8 s-values x 16 k-slots (4 waves).
//     Phase 1: per-thread feature build (GN + leaky 0.2) -> LDS [64][FSTRIDE]
//     Phase 2: hidden = att_w1(32x64) @ feature via V_WMMA_F32_16X16X4_F32;
//              each wave owns 2 N-tiles (its 32 slots), 2 M-tiles, 16 k-steps.
//              Epilogue folds leaky(0.2) + att_w2 dot across the D layout:
//              lane partial over its 8 rows/tile, shfl_xor(16) adds the
//              complementary half-wave, then width-16 softmax butterflies
//              (an N-tile's 16 columns are exactly the 16 k of one s).
//     Phase 3: pooled[c][s] = sum_k att*feat from LDS.
// ---------------------------------------------------------------------------
#define FSLOTS  128            // slots per block = 8 s * 16 k
#define FSTRIDE 130            // padded slot stride (bank-conflict avoidance)
__global__ __launch_bounds__(FSLOTS)
void finalize_kernel(const float* __restrict__ x, const int* __restrict__ idx,
                     const float* __restrict__ stats,
                     const float* __restrict__ conv_w, const float* __restrict__ pos_w,
                     const float* __restrict__ bn_g,  const float* __restrict__ bn_b,
                     const float* __restrict__ bn2_g, const float* __restrict__ bn2_b,
                     const float* __restrict__ att_w1, const float* __restrict__ att_w2,
                     float* __restrict__ pooled, float* __restrict__ out) {
    __shared__ float feat_lds[64 * FSTRIDE];   // [channel][slot]
    __shared__ float w1[32 * 64];              // att_w1 row-major
    __shared__ float wconv[128];
    __shared__ float wpos[320];
    __shared__ float w2s[32];
    __shared__ float gscale[64];
    __shared__ float gshift[64];
    __shared__ float att_lds[FSLOTS];

    const int b   = blockIdx.y;
    const int tid = threadIdx.x;               // == slot
    const int s_local = tid >> 4;              // 0..7
    const int k       = tid & 15;
    const int s   = blockIdx.x * 8 + s_local;

    for (int j = tid; j < 128;  j += FSLOTS) wconv[j] = conv_w[j];
    for (int j = tid; j < 320;  j += FSLOTS) wpos[j]  = pos_w[j];
    for (int j = tid; j < 2048; j += FSLOTS) w1[j]    = att_w1[j];
    if (tid < 32) w2s[tid] = att_w2[tid];
    if (tid < 64) {
        const float N = (float)(NS * KNN);
        float sum = stats[(b * 64 + tid) * 2 + 0];
        float sq  = stats[(b * 64 + tid) * 2 + 1];
        float mu  = sum / N;
        float var = sq / N - mu * mu;
        float rstd = rsqrtf(var + 1e-5f);
        float g  = (tid < 32) ? bn_g[tid] : bn2_g[tid - 32];
        float bb = (tid < 32) ? bn_b[tid] : bn2_b[tid - 32];
        gscale[tid] = g * rstd;
        gshift[tid] = bb - mu * g * rstd;
    }
    __syncthreads();

    // ---- Phase 1: build feature[64] for this slot, write to LDS ----
    const float* xb = x + (size_t)b * CCH * NP;
    const int*   ib = idx + ((size_t)(b * NS + s)) * KNN;
    const int pk = ib[k];
    const int p0 = ib[0];

    float x1c[4];
#pragma unroll
    for (int c = 0; c < 4; ++c) x1c[c] = xb[c * NP + pk];
    float pc0 = xb[p0], pc1 = xb[NP + p0], pc2 = xb[2 * NP + p0];
    float d0 = x1c[0] - pc0, d1 = x1c[1] - pc1, d2 = x1c[2] - pc2;
    float pemb[10] = { x1c[0], x1c[1], x1c[2], pc0, pc1, pc2,
                       d0, d1, d2, sqrtf(d0 * d0 + d1 * d1 + d2 * d2) };

#pragma unroll 4
    for (int o = 0; o < 32; ++o) {
        float y = 0.0f;
#pragma unroll
        for (int c = 0; c < 4; ++c) y += wconv[o * 4 + c] * x1c[c];
        y = y * gscale[o] + gshift[o];
        feat_lds[o * FSTRIDE + tid] = leakyf(y, 0.2f);
    }
#pragma unroll 2
    for (int o = 0; o < 32; ++o) {
        float y = 0.0f;
#pragma unroll
        for (int c = 0; c < 10; ++c) y += wpos[o * 10 + c] * pemb[c];
        y = y * gscale[32 + o] + gshift[32 + o];
        feat_lds[(32 + o) * FSTRIDE + tid] = leakyf(y, 0.2f);
    }

    // output channels 0..3 = raw features of nearest neighbor (k == 0 slot)
    if (k == 0) {
#pragma unroll
        for (int c = 0; c < 4; ++c)
            out[((size_t)b * 68 + c) * NS + s] = x1c[c];
    }
    __syncthreads();

    // ---- Phase 2: WMMA hidden layer + attention logits + softmax ----
    const int wave = tid >> 5;                 // 0..3
    const int lane = tid & 31;
    const int n    = lane & 15;
    const int krow = (lane < 16) ? 0 : 2;      // K-pair owned by this half-wave
    const int rowoff = (lane >> 4) * 8;        // hi half-wave holds M = j + 8

#pragma unroll
    for (int t2 = 0; t2 < 2; ++t2) {
        const int tile = wave * 2 + t2;        // N-tile == one s (16 k columns)
        const int col  = tile * 16 + n;        // slot index of this lane's column

        v8f acc[2];
#pragma unroll
        for (int m = 0; m < 2; ++m) acc[m] = (v8f){};

#pragma unroll
        for (int kk = 0; kk < 16; ++kk) {
            v2f bf;
            bf.x = feat_lds[(kk * 4 + krow + 0) * FSTRIDE + col];
            bf.y = feat_lds[(kk * 4 + krow + 1) * FSTRIDE + col];
#pragma unroll
            for (int m = 0; m < 2; ++m) {
                const int row = m * 16 + n;
                v2f af;
                af.x = w1[row * 64 + kk * 4 + krow + 0];
                af.y = w1[row * 64 + kk * 4 + krow + 1];
                acc[m] = __builtin_amdgcn_wmma_f32_16x16x4_f32(
                    false, af, false, bf, (short)0, acc[m], false, false);
            }
        }

        // lane partial of att_w2 . leaky(hidden) over its 16 of 32 rows
        float partial = 0.0f;
#pragma unroll
        for (int m = 0; m < 2; ++m) {
#pragma unroll
            for (int j = 0; j < 8; ++j) {
                const int o = m * 16 + j + rowoff;
                partial += w2s[o] * leakyf(acc[m][j], 0.2f);
            }
        }
        float logit = partial + __shfl_xor(partial, 16, 32);

        // softmax over the tile's 16 columns (= 16 k of one s)
        float mx = logit;
        mx = fmaxf(mx, __shfl_xor(mx, 1, 16));
        mx = fmaxf(mx, __shfl_xor(mx, 2, 16));
        mx = fmaxf(mx, __shfl_xor(mx, 4, 16));
        mx = fmaxf(mx, __shfl_xor(mx, 8, 16));
        float e = __expf(logit - mx);
        float ssum = e;
        ssum += __shfl_xor(ssum, 1, 16);
        ssum += __shfl_xor(ssum, 2, 16);
        ssum += __shfl_xor(ssum, 4, 16);
        ssum += __shfl_xor(ssum, 8, 16);
        if (lane < 16) att_lds[col] = e / ssum;
    }
    __syncthreads();

    // ---- Phase 3: pooled[c][s] = sum_k att[k] * feat[c][k] ----
    float* pb = pooled + ((size_t)b * 64) * NS;
#pragma unroll
    for (int i = 0; i < 4; ++i) {
        const int v  = i * FSLOTS + tid;       // 0..511
        const int c  = v >> 3;                 // 0..63
        const int sl = v & 7;                  // 0..7
        float acc = 0.0f;
#pragma unroll
        for (int kk = 0; kk < 16; ++kk)
            acc += feat_lds[c * FSTRIDE + sl * 16 + kk] * att_lds[sl * 16 + kk];
        pb[(size_t)c * NS + (blockIdx.x * 8 + sl)] = acc;
    }
}

// ---------------------------------------------------------------------------
// K4: out[4..67] = leaky( b1_w(64x64) @ pooled + b2_w(64x4) @ x1[...,0], 0.1 )
//     via V_WMMA_F32_16X16X4_F32: K=64 -> 16 chained k-steps, M=64 -> 4 row
//     tiles, each wave owns an N=16 column tile (16 s-values).
// ---------------------------------------------------------------------------
__global__ __launch_bounds__(256)
void out_gemm_kernel(const float* __restrict__ pooled,
                     const float* __restrict__ b1_w, const float* __restrict__ b2_w,
                     const float* __restrict__ x, const int* __restrict__ idx,
                     float* __restrict__ out) {
    __shared__ float w[64 * 64];    // b1_w row-major [o][c]
    __shared__ float w2s[64 * 4];
    const int b    = blockIdx.y;
    const int tid  = threadIdx.x;
    for (int j = tid; j < 4096; j += 256) w[j]   = b1_w[j];
    for (int j = tid; j < 256;  j += 256) w2s[j] = b2_w[j];
    __syncthreads();

    const int wave = tid >> 5;            // 8 waves -> 128 s per block
    const int lane = tid & 31;
    const int n    = lane & 15;
    const int s    = blockIdx.x * 128 + wave * 16 + n;
    const int krow = (lane < 16) ? 0 : 2; // K-pair owned by this half-wave

    const float* pb = pooled + ((size_t)b * 64) * NS;

    v8f acc[4];
#pragma unroll
    for (int m = 0; m < 4; ++m) acc[m] = (v8f){};

    for (int kk = 0; kk < 16; ++kk) {
        v2f bf;
        bf.x = pb[(size_t)(kk * 4 + krow + 0) * NS + s];
        bf.y = pb[(size_t)(kk * 4 + krow + 1) * NS + s];
#pragma unroll
        for (int m = 0; m < 4; ++m) {
            int row = m * 16 + n;
            v2f af;
            af.x = w[row * 64 + kk * 4 + krow + 0];
            af.y = w[row * 64 + kk * 4 + krow + 1];
            acc[m] = __builtin_amdgcn_wmma_f32_16x16x4_f32(
                false, af, false, bf, (short)0, acc[m], false, false);
        }
    }

    // epilogue: + b2_w @ x1[:, :, s, 0], leaky 0.1, store channels 4..67
    const float* xb = x + (size_t)b * CCH * NP;
    const int p0 = idx[((size_t)(b * NS + s)) * KNN];
    float x1c[4];
#pragma unroll
    for (int c = 0; c < 4; ++c) x1c[c] = xb[c * NP + p0];

    const int rowoff = (lane >> 4) * 8;   // hi half-wave holds M = j + 8
#pragma unroll
    for (int m = 0; m < 4; ++m) {
#pragma unroll
        for (int j = 0; j < 8; ++j) {
            int o = m * 16 + j + rowoff;
            float f2 = w2s[o * 4 + 0] * x1c[0] + w2s[o * 4 + 1] * x1c[1]
                     + w2s[o * 4 + 2] * x1c[2] + w2s[o * 4 + 3] * x1c[3];
            float v = leakyf(acc[m][j] + f2, 0.1f);
            out[((size_t)b * 68 + 4 + o) * NS + s] = v;
        }
    }
}

// ---------------------------------------------------------------------------
// launch
// ---------------------------------------------------------------------------
extern "C" void kernel_launch(void* const* d_in, const int* in_sizes, int n_in,
                              void* d_out, int out_size, void* d_ws, size_t ws_size,
                              hipStream_t stream) {
    const float* x      = (const float*)d_in[0];
    const int*   perm   = (const int*)  d_in[1];
    (void)d_in[2]; // num_points == NS (fixed)
    const float* pos_w  = (const float*)d_in[3];
    const float* bn2_g  = (const float*)d_in[4];
    const float* bn2_b  = (const float*)d_in[5];
    const float* conv_w = (const float*)d_in[6];
    const float* bn_g   = (const float*)d_in[7];
    const float* bn_b   = (const float*)d_in[8];
    const float* att_w1 = (const float*)d_in[9];
    const float* att_w2 = (const float*)d_in[10];
    const float* b1_w   = (const float*)d_in[11];
    const float* b2_w   = (const float*)d_in[12];
    float* out = (float*)d_out;

    char*  ws     = (char*)d_ws;
    int*   idx    = (int*)  (ws + OFF_IDX);
    float* stats  = (float*)(ws + OFF_STATS);
    float* pooled = (float*)(ws + OFF_POOLED);

    hipMemsetAsync(stats, 0, BATCH * 64 * 2 * sizeof(float), stream);

    knn_kernel<<<dim3(NS / 256, BATCH), 256, 0, stream>>>(x, perm, idx);
    stats_kernel<<<dim3(NS * KNN / ST_SLOTS, BATCH), ST_SLOTS, 0, stream>>>(
        x, idx, conv_w, pos_w, stats);
    finalize_kernel<<<dim3(NS / 8, BATCH), FSLOTS, 0, stream>>>(
        x, idx, stats, conv_w, pos_w, bn_g, bn_b, bn2_g, bn2_b,
        att_w1, att_w2, pooled, out);
    out_gemm_kernel<<<dim3(NS / 128, BATCH), 256, 0, stream>>>(
        pooled, b1_w, b2_w, x, idx, out);
}